// LightGCN_55061480734870
// MI455X (gfx1250) — compile-verified
//
#include <hip/hip_runtime.h>
#include <hip/hip_bf16.h>

#define DIM 64

typedef float v2f __attribute__((ext_vector_type(2)));
typedef float v8f __attribute__((ext_vector_type(8)));

// ---------------------------------------------------------------------------
// Native no-return fp32 atomic add at device scope (combines in L2).
// ---------------------------------------------------------------------------
__device__ __forceinline__ void atom_add_f32_dev(float* p, float v) {
    asm volatile("global_atomic_add_f32 %0, %1, off scope:SCOPE_DEV"
                 :: "v"((unsigned long long)(uintptr_t)p), "v"(v)
                 : "memory");
}

// ---------------------------------------------------------------------------
// Zero a buffer (float4 granular).
// ---------------------------------------------------------------------------
__global__ void lgcn_zero_kernel(float* __restrict__ buf, long n_float4) {
    long gid = (long)blockIdx.x * blockDim.x + threadIdx.x;
    if (gid >= n_float4) return;
    ((float4*)buf)[gid] = make_float4(0.f, 0.f, 0.f, 0.f);
}

// ---------------------------------------------------------------------------
// Layer-1 scatter: nxt[dst] += w * e0[src], e0 read directly from the two
// input embedding tables.  16 lanes per edge: lane loads one float4 (b128)
// of the 256B source row, then 4 no-return global_atomic_add_f32.
// Edge streams are read once per layer -> non-temporal (TH_LOAD_NT) so the
// 240MB/layer stream does not evict the L2-resident embedding buffers.
// ---------------------------------------------------------------------------
__global__ void lgcn_scatter_first_kernel(const float* __restrict__ user_emb,
                                          const float* __restrict__ item_emb,
                                          float*       __restrict__ nxt,
                                          const float* __restrict__ ew,
                                          const int*   __restrict__ esrc,
                                          const int*   __restrict__ edst,
                                          long n_edges, int n_users) {
    long gid = (long)blockIdx.x * blockDim.x + threadIdx.x;
    long e   = gid >> 4;
    if (e >= n_edges) return;
    int lane = (int)(gid & 15);
    int   s = __builtin_nontemporal_load(esrc + e);
    int   d = __builtin_nontemporal_load(edst + e);
    float w = __builtin_nontemporal_load(ew + e);
    const float* srow = (s < n_users) ? (user_emb + (size_t)s * DIM)
                                      : (item_emb + (size_t)(s - n_users) * DIM);
    float4 v = ((const float4*)srow)[lane];          // RT: hot in L2
    float* dp = nxt + (size_t)d * DIM + lane * 4;
    atom_add_f32_dev(dp + 0, w * v.x);
    atom_add_f32_dev(dp + 1, w * v.y);
    atom_add_f32_dev(dp + 2, w * v.z);
    atom_add_f32_dev(dp + 3, w * v.w);
}

// ---------------------------------------------------------------------------
// Generic scatter: nxt[dst] += w * cur[src] with cur a contiguous [N,64] buf.
// ---------------------------------------------------------------------------
__global__ void lgcn_scatter_kernel(const float* __restrict__ cur,
                                    float*       __restrict__ nxt,
                                    const float* __restrict__ ew,
                                    const int*   __restrict__ esrc,
                                    const int*   __restrict__ edst,
                                    long n_edges) {
    long gid = (long)blockIdx.x * blockDim.x + threadIdx.x;
    long e   = gid >> 4;
    if (e >= n_edges) return;
    int lane = (int)(gid & 15);
    int   s = __builtin_nontemporal_load(esrc + e);
    int   d = __builtin_nontemporal_load(edst + e);
    float w = __builtin_nontemporal_load(ew + e);
    float4 v = ((const float4*)(cur + (size_t)s * DIM))[lane];   // RT: hot in L2
    float* dp = nxt + (size_t)d * DIM + lane * 4;
    atom_add_f32_dev(dp + 0, w * v.x);
    atom_add_f32_dev(dp + 1, w * v.y);
    atom_add_f32_dev(dp + 2, w * v.z);
    atom_add_f32_dev(dp + 3, w * v.w);
}

// ---------------------------------------------------------------------------
// Batched dot with layer-mean fused:
//   out[b] = ((e0+e1+e2+e3)[u_b]/4) . ((e0+e1+e2+e3)[n_users+i_b]/4)
// One wave32 per 16 pairs.  D = U x I^T via V_WMMA_F32_16X16X4_F32 over K=64.
//
// A 16x4 f32 layout (ISA 7.12.2): lanes 0-15 hold K=0,1 for M=lane; lanes
// 16-31 hold K=2,3 for M=lane-16.  B 4x16 mirrors (N across lanes), and since
// B[k][n] = I[n][k], every lane reads row (lane&15) at cols kb+2*(lane>>4)+{0,1}
// of U (for A) and of I (for B).
// C/D 16x16 f32: vgpr j: lanes 0-15 -> M=j, N=lane; lanes 16-31 -> M=j+8,
// N=lane-16.  Diagonal m: lane m (m<8) holds c[m]; lane m+16 holds c[m-8].
// ---------------------------------------------------------------------------
__global__ __launch_bounds__(256) void lgcn_dot_wmma_kernel(
        const float* __restrict__ user_emb,
        const float* __restrict__ item_emb,
        const float* __restrict__ e1,
        const float* __restrict__ e2,
        const float* __restrict__ e3,
        const int*   __restrict__ users,
        const int*   __restrict__ items,
        float*       __restrict__ out,
        int n_users) {
    int lane = threadIdx.x & 31;
    int tile = blockIdx.x * (blockDim.x >> 5) + (threadIdx.x >> 5);
    int r  = lane & 15;                 // row of the 16-row tile this lane feeds
    int kh = (lane >> 4) << 1;          // K sub-offset: 0 or 2
    int b  = tile * 16 + r;

    int    urow = __builtin_nontemporal_load(users + b);
    int    irow = __builtin_nontemporal_load(items + b);
    size_t unode = (size_t)urow;                    // users are nodes [0, n_users)
    size_t inode = (size_t)n_users + (size_t)irow;  // items are nodes [n_users, ...)

    const float* u0 = user_emb + (size_t)urow * DIM;
    const float* i0 = item_emb + (size_t)irow * DIM;
    const float* u1 = e1 + unode * DIM;  const float* v1 = e1 + inode * DIM;
    const float* u2 = e2 + unode * DIM;  const float* v2 = e2 + inode * DIM;
    const float* u3 = e3 + unode * DIM;  const float* v3 = e3 + inode * DIM;

    v8f c = {};
#pragma unroll
    for (int kb = 0; kb < DIM; kb += 4) {
        int o = kb + kh;
        v2f a, bm;
        a.x  = u0[o]     + u1[o]     + u2[o]     + u3[o];
        a.y  = u0[o + 1] + u1[o + 1] + u2[o + 1] + u3[o + 1];
        bm.x = i0[o]     + v1[o]     + v2[o]     + v3[o];
        bm.y = i0[o + 1] + v1[o + 1] + v2[o + 1] + v3[o + 1];
        c = __builtin_amdgcn_wmma_f32_16x16x4_f32(
                /*neg_a=*/false, a, /*neg_b=*/false, bm,
                /*c_mod=*/(short)0, c, /*reuse_a=*/false, /*reuse_b=*/false);
    }

    // Diagonal extraction + combined (1/4)*(1/4) layer-mean scaling.
#pragma unroll
    for (int j = 0; j < 8; ++j) {
        int m = j + ((lane >> 4) << 3);     // M held by this lane in vgpr j
        if ((lane & 15) == m) {             // diagonal: N == M
            out[tile * 16 + m] = c[j] * 0.0625f;
        }
    }
}

// ---------------------------------------------------------------------------
// Host-side orchestration (deterministic, graph-capture safe).
// ---------------------------------------------------------------------------
extern "C" void kernel_launch(void* const* d_in, const int* in_sizes, int n_in,
                              void* d_out, int out_size, void* d_ws, size_t ws_size,
                              hipStream_t stream) {
    const float* user_emb = (const float*)d_in[0];
    const float* item_emb = (const float*)d_in[1];
    const float* edge_w   = (const float*)d_in[2];
    const int*   edge_src = (const int*)d_in[3];
    const int*   edge_dst = (const int*)d_in[4];
    const int*   users    = (const int*)d_in[5];
    const int*   items    = (const int*)d_in[6];
    float*       out      = (float*)d_out;

    const int  n_users = in_sizes[0] / DIM;
    const int  n_items = in_sizes[1] / DIM;
    const int  n_nodes = n_users + n_items;
    const long n_edges = (long)in_sizes[2];
    const int  batch   = in_sizes[5];
    (void)ws_size; (void)n_in; (void)out_size;

    const size_t node_floats = (size_t)n_nodes * DIM;
    float* e1 = (float*)d_ws;            // layer-1 output
    float* e2 = e1 + node_floats;        // layer-2 output
    float* e3 = e2 + node_floats;        // layer-3 output

    const long nf4 = (long)n_nodes * (DIM / 4);
    const int  T   = 256;
    const unsigned zero_blocks    = (unsigned)((nf4 + T - 1) / T);
    const unsigned scatter_blocks = (unsigned)((n_edges * 16 + T - 1) / T);

    // Layer 1: e1 = SpMM(e0) with e0 read from the input tables.
    lgcn_zero_kernel<<<zero_blocks, T, 0, stream>>>(e1, nf4);
    lgcn_scatter_first_kernel<<<scatter_blocks, T, 0, stream>>>(
        user_emb, item_emb, e1, edge_w, edge_src, edge_dst, n_edges, n_users);

    // Layer 2: e2 = SpMM(e1)
    lgcn_zero_kernel<<<zero_blocks, T, 0, stream>>>(e2, nf4);
    lgcn_scatter_kernel<<<scatter_blocks, T, 0, stream>>>(
        e1, e2, edge_w, edge_src, edge_dst, n_edges);

    // Layer 3: e3 = SpMM(e2)
    lgcn_zero_kernel<<<zero_blocks, T, 0, stream>>>(e3, nf4);
    lgcn_scatter_kernel<<<scatter_blocks, T, 0, stream>>>(
        e2, e3, edge_w, edge_src, edge_dst, n_edges);

    // Fused layer-mean + batched pair dots via WMMA: 16 pairs/wave, 8 waves/block.
    const int tiles = batch / 16;        // 1024
    lgcn_dot_wmma_kernel<<<tiles / 8, 256, 0, stream>>>(
        user_emb, item_emb, e1, e2, e3, users, items, out, n_users);
}